// ParallelMultiScaleLTCModel_38439957299384
// MI455X (gfx1250) — compile-verified
//
#include <hip/hip_runtime.h>

#define Hd   30
#define HP   32
#define IND  2
#define OUTD 2
#define BT   4096
#define TT   512
#define NW   4
#define TPB  (NW * 32)
#define RS   132   // padded row stride for h buffer (132 mod 64 = 4 -> conflict-free rows)

typedef float v8f __attribute__((ext_vector_type(8)));
typedef float v2f __attribute__((ext_vector_type(2)));

__device__ __forceinline__ v8f wmma4(v2f a, v2f b, v8f c) {
  // D = A(16x4 f32) * B(4x16 f32) + C(16x16 f32)
  return __builtin_amdgcn_wmma_f32_16x16x4_f32(false, a, false, b, (short)0, c, false, false);
}

// LTC elementwise update, division-free:
//   s       = 1/(1+exp(-cm))                       (v_rcp)
//   h/dtc   = h * clamp(1/tau + s, 1, 10)          (dtc clamp [0.1,1] inverted)
//   dx      = s*cm - h/dtc ; hnew = h + dx
__device__ __forceinline__ void ltc_elem(float rtau, v8f cmb, v8f hold, v8f& hnew, v8f& dx) {
#pragma unroll
  for (int v = 0; v < 8; ++v) {
    float cm   = cmb[v];
    float s    = __builtin_amdgcn_rcpf(1.f + __expf(-cm));
    float idtc = fminf(fmaxf(rtau + s, 1.0f), 10.0f);
    float d    = s * cm - hold[v] * idtc;
    dx[v]      = d;
    hnew[v]    = hold[v] + d;
  }
}

// stage Wh (H x H row-major) as B-operand, K-pair packed: dst[kp][n] = (Wh[n][2kp], Wh[n][2kp+1])
__device__ __forceinline__ void stageWh(float (*dst)[HP][2], const float* W, int tid) {
  for (int i = tid; i < 16 * HP; i += TPB) {
    int kp = i / HP, n = i % HP;
    float a = 0.f, b = 0.f;
    if (n < Hd) {
      int k0 = 2 * kp;
      if (k0     < Hd) a = W[n * Hd + k0];
      if (k0 + 1 < Hd) b = W[n * Hd + k0 + 1];
    }
    dst[kp][n][0] = a; dst[kp][n][1] = b;
  }
}

__device__ __forceinline__ void stageWi(float (*dst)[HP][2], const float* W, int tid) {
  for (int i = tid; i < 2 * HP; i += TPB) {
    int kp = i / HP, n = i % HP;
    float a = 0.f, b = 0.f;
    if (kp == 0 && n < Hd) { a = W[n * IND + 0]; b = W[n * IND + 1]; }
    dst[kp][n][0] = a; dst[kp][n][1] = b;
  }
}

__device__ __forceinline__ float w4elem(int k, int n, const float* Wi4, const float* Wh4) {
  if (n >= Hd) return 0.f;
  if (k < 96) { int c = k >> 5, j = k & 31; return (j < Hd) ? Wi4[n * (3 * Hd) + c * Hd + j] : 0.f; }
  int j = k - 96;
  return (j < Hd) ? Wh4[n * Hd + j] : 0.f;
}

__launch_bounds__(TPB)
__global__ void ltc_wmma_kernel(
    const float* __restrict__ x,
    const float* __restrict__ Wi1, const float* __restrict__ bi1,
    const float* __restrict__ Wh1, const float* __restrict__ bh1,
    const float* __restrict__ Wi2, const float* __restrict__ bi2,
    const float* __restrict__ Wh2, const float* __restrict__ bh2,
    const float* __restrict__ Wi3, const float* __restrict__ bi3,
    const float* __restrict__ Wh3, const float* __restrict__ bh3,
    const float* __restrict__ Wi4, const float* __restrict__ bi4,
    const float* __restrict__ Wh4, const float* __restrict__ bh4,
    const float* __restrict__ Wf,  const float* __restrict__ bf,
    float* __restrict__ out)
{
  __shared__ __attribute__((aligned(16))) float sWh[3][16][HP][2]; // recurrent weights cells 1-3
  __shared__ __attribute__((aligned(16))) float sW4[64][HP][2];    // cell4: [Wi4 over 96K | Wh4 over 32K]
  __shared__ __attribute__((aligned(16))) float sWi[3][2][HP][2];  // input weights cells 1-3 (K padded to 4)
  __shared__ __attribute__((aligned(16))) float sBias[4][HP];      // bi + bh, padded
  __shared__ __attribute__((aligned(16))) float sWf[OUTD * Hd];
  __shared__ __attribute__((aligned(16))) float sBf[OUTD];
  __shared__ __attribute__((aligned(16))) float sH[NW][16][RS];    // per-wave [row][h1|h2|h3|h4]

  const int tid = threadIdx.x;

  for (int i = tid; i < NW * 16 * RS; i += TPB) ((float*)sH)[i] = 0.f;

  stageWh(sWh[0], Wh1, tid);
  stageWh(sWh[1], Wh2, tid);
  stageWh(sWh[2], Wh3, tid);
  stageWi(sWi[0], Wi1, tid);
  stageWi(sWi[1], Wi2, tid);
  stageWi(sWi[2], Wi3, tid);
  for (int i = tid; i < 64 * HP; i += TPB) {
    int kp = i / HP, n = i % HP;
    sW4[kp][n][0] = w4elem(2 * kp,     n, Wi4, Wh4);
    sW4[kp][n][1] = w4elem(2 * kp + 1, n, Wi4, Wh4);
  }
  for (int i = tid; i < 4 * HP; i += TPB) {
    int c = i / HP, n = i % HP;
    const float* bis[4] = {bi1, bi2, bi3, bi4};
    const float* bhs[4] = {bh1, bh2, bh3, bh4};
    sBias[c][n] = (n < Hd) ? (bis[c][n] + bhs[c][n]) : 0.f;
  }
  for (int i = tid; i < OUTD * Hd; i += TPB) sWf[i] = Wf[i];
  if (tid < OUTD) sBf[tid] = bf[tid];
  __syncthreads();

  const int lane = tid & 31;
  const int wave = tid >> 5;
  const int row  = lane & 15;   // M row (A) / N col (B,C,D)
  const int hi   = lane >> 4;   // lane-half selects K pair (A,B) / M+8 (C,D)
  const int b0w  = blockIdx.x * (NW * 16) + wave * 16;

  v8f hreg[4][2];
#pragma unroll
  for (int c = 0; c < 4; ++c)
#pragma unroll
    for (int nh = 0; nh < 2; ++nh)
#pragma unroll
      for (int v = 0; v < 8; ++v) hreg[c][nh][v] = 0.f;

  const size_t dxstep = (size_t)BT * Hd;            // one timestep of one dx tensor
  float* dx0 = out + (size_t)BT * OUTD;             // dx tensors follow `out` in the flat tuple

  for (int t = 0; t < TT; ++t) {
    // x A-operand: lanes 0-15 hold (x0,x1); lanes 16-31 hold K=2,3 zero pad
    v2f xa; xa.x = 0.f; xa.y = 0.f;
    if (hi == 0) {
      const float* xp = x + ((size_t)(b0w + row) * TT + t) * IND;
      xa.x = xp[0]; xa.y = xp[1];
    }

    // ---- cells 1..3 (use OLD h, independent) ----
#pragma unroll
    for (int c = 0; c < 3; ++c) {
      const float rtau = (c == 0) ? (1.0f / 0.7f) : ((c == 2) ? 0.1f : 1.0f);
#pragma unroll
      for (int nh = 0; nh < 2; ++nh) {
        const int n = nh * 16 + row;
        const float bv0 = sBias[c][n];
        v8f acc = {bv0, bv0, bv0, bv0, bv0, bv0, bv0, bv0};
        acc = wmma4(xa, *(const v2f*)&sWi[c][hi][n][0], acc);
#pragma unroll
        for (int k = 0; k < 8; ++k) {
          v2f av = *(const v2f*)&sH[wave][row][c * HP + 4 * k + 2 * hi];
          v2f bvv = *(const v2f*)&sWh[c][2 * k + hi][n][0];
          acc = wmma4(av, bvv, acc);
        }
        v8f hnew, dxv;
        ltc_elem(rtau, acc, hreg[c][nh], hnew, dxv);
        hreg[c][nh] = hnew;
#pragma unroll
        for (int v = 0; v < 8; ++v) sH[wave][v + 8 * hi][c * HP + n] = hnew[v];
        if (n < Hd) {
          float* dst = dx0 + (size_t)c * TT * dxstep + (size_t)t * dxstep + (size_t)b0w * Hd + n;
#pragma unroll
          for (int v = 0; v < 8; ++v) dst[(size_t)(v + 8 * hi) * Hd] = dxv[v];
        }
      }
    }
    __syncthreads();

    // ---- cell 4: input = [h1 h2 h3] (NEW) concatenated, recurrent on OLD h4 ----
    {
      const float rtau = 1.0f;
#pragma unroll
      for (int nh = 0; nh < 2; ++nh) {
        const int n = nh * 16 + row;
        const float bv0 = sBias[3][n];
        v8f acc = {bv0, bv0, bv0, bv0, bv0, bv0, bv0, bv0};
#pragma unroll
        for (int k = 0; k < 32; ++k) {
          v2f av = *(const v2f*)&sH[wave][row][4 * k + 2 * hi];
          v2f bvv = *(const v2f*)&sW4[2 * k + hi][n][0];
          acc = wmma4(av, bvv, acc);
        }
        v8f hnew, dxv;
        ltc_elem(rtau, acc, hreg[3][nh], hnew, dxv);
        hreg[3][nh] = hnew;
#pragma unroll
        for (int v = 0; v < 8; ++v) sH[wave][v + 8 * hi][3 * HP + n] = hnew[v];
        if (n < Hd) {
          float* dst = dx0 + (size_t)3 * TT * dxstep + (size_t)t * dxstep + (size_t)b0w * Hd + n;
#pragma unroll
          for (int v = 0; v < 8; ++v) dst[(size_t)(v + 8 * hi) * Hd] = dxv[v];
        }
      }
    }
    __syncthreads();
  }

  // ---- final projection: out[b][o] = h4[b] . Wf[o] + bf[o] ----
  {
    const int o = hi;  // lanes 0-15 -> out col 0, lanes 16-31 -> out col 1
    float a = sBf[o];
#pragma unroll
    for (int j = 0; j < Hd; ++j) a += sH[wave][row][3 * HP + j] * sWf[o * Hd + j];
    out[(size_t)(b0w + row) * OUTD + o] = a;
  }
}

extern "C" void kernel_launch(void* const* d_in, const int* in_sizes, int n_in,
                              void* d_out, int out_size, void* d_ws, size_t ws_size,
                              hipStream_t stream) {
  (void)in_sizes; (void)n_in; (void)out_size; (void)d_ws; (void)ws_size;
  const float* p[19];
  for (int i = 0; i < 19; ++i) p[i] = (const float*)d_in[i];
  dim3 grid(BT / (NW * 16));  // 64 blocks
  dim3 block(TPB);            // 128 threads = 4 waves
  hipLaunchKernelGGL(ltc_wmma_kernel, grid, block, 0, stream,
                     p[0],
                     p[1],  p[2],  p[3],  p[4],
                     p[5],  p[6],  p[7],  p[8],
                     p[9],  p[10], p[11], p[12],
                     p[13], p[14], p[15], p[16],
                     p[17], p[18],
                     (float*)d_out);
}